// Surrogate_26517128085853
// MI455X (gfx1250) — compile-verified
//
#include <hip/hip_runtime.h>
#include <hip/hip_bf16.h>

#define DD 64
#define EPSV 1e-5f

typedef float v2f __attribute__((ext_vector_type(2)));
typedef float v8f __attribute__((ext_vector_type(8)));

// ---------------------------------------------------------------------------
// MLP: out = relu(x @ W1 + b1) @ W2 + b2, D=64, using V_WMMA_F32_16X16X4_F32
// 8 waves per block; each wave computes a 16-row tile of the 64-wide output.
// ---------------------------------------------------------------------------
__global__ __launch_bounds__(256) void mlp_wmma_kernel(
    const float* __restrict__ xin, int nrows,
    const float* __restrict__ W1, const float* __restrict__ b1,
    const float* __restrict__ W2, const float* __restrict__ b2,
    float* __restrict__ out)
{
    __shared__ float sW1[DD * DD];
    __shared__ float sW2[DD * DD];
    __shared__ float sB1[DD];
    __shared__ float sB2[DD];
    __shared__ float sTile[8][16 * DD];  // per-wave A/H scratch

    const int tid = threadIdx.x;
    for (int i = tid; i < DD * DD; i += 256) { sW1[i] = W1[i]; sW2[i] = W2[i]; }
    if (tid < DD) { sB1[tid] = b1[tid]; sB2[tid] = b2[tid]; }
    __syncthreads();

    const int wave = tid >> 5;
    const int lane = tid & 31;
    const int row0 = (blockIdx.x * 8 + wave) * 16;
    if (row0 >= nrows) return;

    float* tile = sTile[wave];

    // Stage 16x64 input tile into LDS (zero-pad tail rows); coalesced.
    for (int i = lane; i < 16 * DD; i += 32) {
        int r = row0 + (i >> 6);
        tile[i] = (r < nrows) ? xin[r * DD + (i & 63)] : 0.0f;
    }

    const int laneM  = lane & 15;  // M for A / N for B,C,D
    const int laneHi = lane >> 4;  // 0: K+{0,1} (or M 0..7), 1: K+{2,3} (or M 8..15)
    const v8f vzero = {0.f, 0.f, 0.f, 0.f, 0.f, 0.f, 0.f, 0.f};

    v8f acc[4];

    // ---- GEMM1: H = A @ W1 ----
    for (int t = 0; t < 4; ++t) acc[t] = vzero;
    for (int k = 0; k < 16; ++k) {
        const int k0 = k * 4 + laneHi * 2;
        v2f a; a.x = tile[laneM * DD + k0]; a.y = tile[laneM * DD + k0 + 1];
        for (int t = 0; t < 4; ++t) {
            const int n0 = t * 16;
            v2f b; b.x = sW1[k0 * DD + n0 + laneM];
                   b.y = sW1[(k0 + 1) * DD + n0 + laneM];
            acc[t] = __builtin_amdgcn_wmma_f32_16x16x4_f32(
                false, a, false, b, (short)0, acc[t], false, false);
        }
    }
    // bias + ReLU, write hidden tile back to LDS in row-major (D-layout unpack)
    for (int t = 0; t < 4; ++t) {
        const int n = t * 16 + laneM;
        const float bias = sB1[n];
        for (int j = 0; j < 8; ++j) {
            const int m = j + laneHi * 8;
            tile[m * DD + n] = fmaxf(acc[t][j] + bias, 0.0f);
        }
    }

    // ---- GEMM2: O = H @ W2 ----
    for (int t = 0; t < 4; ++t) acc[t] = vzero;
    for (int k = 0; k < 16; ++k) {
        const int k0 = k * 4 + laneHi * 2;
        v2f a; a.x = tile[laneM * DD + k0]; a.y = tile[laneM * DD + k0 + 1];
        for (int t = 0; t < 4; ++t) {
            const int n0 = t * 16;
            v2f b; b.x = sW2[k0 * DD + n0 + laneM];
                   b.y = sW2[(k0 + 1) * DD + n0 + laneM];
            acc[t] = __builtin_amdgcn_wmma_f32_16x16x4_f32(
                false, a, false, b, (short)0, acc[t], false, false);
        }
    }
    for (int t = 0; t < 4; ++t) {
        const int n = t * 16 + laneM;
        const float bias = sB2[n];
        for (int j = 0; j < 8; ++j) {
            const int m = j + laneHi * 8;
            const int r = row0 + m;
            if (r < nrows) out[r * DD + n] = acc[t][j] + bias;
        }
    }
}

// ---------------------------------------------------------------------------
// Pass 1: per-row LayerNorm (recomputed), accumulate column sum/sumsq of z.
// One wave per row (grid-stride). Block-local LDS accum -> global atomics.
// ---------------------------------------------------------------------------
__global__ __launch_bounds__(256) void ln_stats_kernel(
    const float* __restrict__ src,
    const float* __restrict__ ln_g, const float* __restrict__ ln_b,
    float* __restrict__ stats, int n)
{
    __shared__ float ssum[DD];
    __shared__ float ssq[DD];
    const int tid = threadIdx.x, lane = tid & 31, wave = tid >> 5;
    if (tid < DD) { ssum[tid] = 0.f; ssq[tid] = 0.f; }
    __syncthreads();

    const float g0 = ln_g[lane], g1 = ln_g[lane + 32];
    const float b0 = ln_b[lane], b1 = ln_b[lane + 32];
    const int nw = gridDim.x * 8;

    for (int row = blockIdx.x * 8 + wave; row < n; row += nw) {
        const float x0 = src[row * DD + lane];
        const float x1 = src[row * DD + lane + 32];
        float s = x0 + x1;
        for (int off = 16; off > 0; off >>= 1) s += __shfl_xor(s, off, 32);
        const float m = s * (1.0f / 64.0f);
        const float d0 = x0 - m, d1 = x1 - m;
        float v = d0 * d0 + d1 * d1;
        for (int off = 16; off > 0; off >>= 1) v += __shfl_xor(v, off, 32);
        const float inv = rsqrtf(v * (1.0f / 64.0f) + EPSV);
        const float z0 = d0 * inv * g0 + b0;
        const float z1 = d1 * inv * g1 + b1;
        atomicAdd(&ssum[lane], z0);        atomicAdd(&ssum[lane + 32], z1);
        atomicAdd(&ssq[lane], z0 * z0);    atomicAdd(&ssq[lane + 32], z1 * z1);
    }
    __syncthreads();
    if (tid < DD) {
        atomicAdd(&stats[tid], ssum[tid]);
        atomicAdd(&stats[DD + tid], ssq[tid]);
    }
}

__global__ void zero_stats_kernel(float* __restrict__ stats)
{
    if (threadIdx.x < 4 * DD) stats[threadIdx.x] = 0.0f;
}

// Fold BatchNorm (biased batch stats) into per-column scale/shift.
__global__ void bn_finalize_kernel(float* __restrict__ stats,
                                   const float* __restrict__ bn_g,
                                   const float* __restrict__ bn_b, int n)
{
    const int c = threadIdx.x;
    if (c < DD) {
        const float invn = 1.0f / (float)n;
        const float mean = stats[c] * invn;
        const float var  = stats[DD + c] * invn - mean * mean;
        const float scl  = rsqrtf(var + EPSV) * bn_g[c];
        stats[2 * DD + c] = scl;
        stats[3 * DD + c] = bn_b[c] - mean * scl;
    }
}

// ---------------------------------------------------------------------------
// Pass 2: recompute LN per row, apply folded BN scale/shift, optional ReLU,
// write dst (in-place safe) and optionally a second copy (x0 stash).
// ---------------------------------------------------------------------------
__global__ __launch_bounds__(256) void ln_bn_apply_kernel(
    const float* __restrict__ src,
    const float* __restrict__ ln_g, const float* __restrict__ ln_b,
    const float* __restrict__ stats,
    float* __restrict__ dst, float* __restrict__ dst2,
    int n, int do_relu)
{
    const int tid = threadIdx.x, lane = tid & 31, wave = tid >> 5;
    const float g0 = ln_g[lane], g1 = ln_g[lane + 32];
    const float b0 = ln_b[lane], b1 = ln_b[lane + 32];
    const float s0 = stats[2 * DD + lane], s1 = stats[2 * DD + lane + 32];
    const float t0 = stats[3 * DD + lane], t1 = stats[3 * DD + lane + 32];
    const int nw = gridDim.x * 8;

    for (int row = blockIdx.x * 8 + wave; row < n; row += nw) {
        const float x0 = src[row * DD + lane];
        const float x1 = src[row * DD + lane + 32];
        float s = x0 + x1;
        for (int off = 16; off > 0; off >>= 1) s += __shfl_xor(s, off, 32);
        const float m = s * (1.0f / 64.0f);
        const float d0 = x0 - m, d1 = x1 - m;
        float v = d0 * d0 + d1 * d1;
        for (int off = 16; off > 0; off >>= 1) v += __shfl_xor(v, off, 32);
        const float inv = rsqrtf(v * (1.0f / 64.0f) + EPSV);
        float y0 = (d0 * inv * g0 + b0) * s0 + t0;
        float y1 = (d1 * inv * g1 + b1) * s1 + t1;
        if (do_relu) { y0 = fmaxf(y0, 0.f); y1 = fmaxf(y1, 0.f); }
        dst[row * DD + lane]      = y0;
        dst[row * DD + lane + 32] = y1;
        if (dst2) { dst2[row * DD + lane] = y0; dst2[row * DD + lane + 32] = y1; }
    }
}

// agg = x0 (float4 streaming copy)
__global__ __launch_bounds__(256) void copy4_kernel(
    const float4* __restrict__ s, float4* __restrict__ d, int n)
{
    int i = blockIdx.x * blockDim.x + threadIdx.x;
    const int stride = gridDim.x * blockDim.x;
    for (; i < n; i += stride) d[i] = s[i];
}

// ---------------------------------------------------------------------------
// SpMM: agg[r] += val * x[c] for each COO edge. One wave per edge.
// Edge stream loaded non-temporally so x/agg stay resident in the 192MB L2.
// ---------------------------------------------------------------------------
__global__ __launch_bounds__(256) void spmm_kernel(
    const int* __restrict__ erow, const int* __restrict__ ecol,
    const float* __restrict__ eval,
    const float* __restrict__ x, float* __restrict__ agg, int nnz)
{
    const int lane = threadIdx.x & 31;
    const int wave = (blockIdx.x * blockDim.x + threadIdx.x) >> 5;
    const int nwaves = (gridDim.x * blockDim.x) >> 5;

    for (int e = wave; e < nnz; e += nwaves) {
        const int   r = __builtin_nontemporal_load(&erow[e]);
        const int   c = __builtin_nontemporal_load(&ecol[e]);
        const float v = __builtin_nontemporal_load(&eval[e]);
        const float2 xv = reinterpret_cast<const float2*>(x + c * DD)[lane];
        atomicAdd(&agg[r * DD + 2 * lane],     v * xv.x);
        atomicAdd(&agg[r * DD + 2 * lane + 1], v * xv.y);
    }
}

// ---------------------------------------------------------------------------
extern "C" void kernel_launch(void* const* d_in, const int* in_sizes, int n_in,
                              void* d_out, int out_size, void* d_ws, size_t ws_size,
                              hipStream_t stream)
{
    const int NU = 200000, NF = 1000, NI = 100000;
    const int N  = NU + NF + NI;            // 301000
    const int NNZ = 10000000;

    const int*   erow   = (const int*)  d_in[0];
    const int*   ecol   = (const int*)  d_in[1];
    const float* evalv  = (const float*)d_in[2];
    const float* user_w = (const float*)d_in[3];
    const float* faker_w= (const float*)d_in[4];
    const float* item_w = (const float*)d_in[5];
    const float* uW1 = (const float*)d_in[6];
    const float* ub1 = (const float*)d_in[7];
    const float* uW2 = (const float*)d_in[8];
    const float* ub2 = (const float*)d_in[9];
    const float* iW1 = (const float*)d_in[10];
    const float* ib1 = (const float*)d_in[11];
    const float* iW2 = (const float*)d_in[12];
    const float* ib2 = (const float*)d_in[13];
    const float* ln_g = (const float*)d_in[14];
    const float* ln_b = (const float*)d_in[15];
    const float* bn_g = (const float*)d_in[16];
    const float* bn_b = (const float*)d_in[17];

    float* X     = (float*)d_out;                 // live features (also final out)
    float* X0    = (float*)d_ws;                  // N*64 floats
    float* AGG   = X0 + (size_t)N * DD;           // N*64 floats
    float* stats = AGG + (size_t)N * DD;          // 256 floats

    // ---- embeddings -> MLP projections (WMMA) ----
    mlp_wmma_kernel<<<(NU + 127) / 128, 256, 0, stream>>>(
        user_w, NU, uW1, ub1, uW2, ub2, X);
    mlp_wmma_kernel<<<(NF + 127) / 128, 256, 0, stream>>>(
        faker_w, NF, uW1, ub1, uW2, ub2, X + (size_t)NU * DD);
    mlp_wmma_kernel<<<(NI + 127) / 128, 256, 0, stream>>>(
        item_w, NI, iW1, ib1, iW2, ib2, X + (size_t)(NU + NF) * DD);

    // ---- stage 0: x = BN(LN(x)); x0 = x ----
    zero_stats_kernel<<<1, 256, 0, stream>>>(stats);
    ln_stats_kernel<<<1024, 256, 0, stream>>>(X, ln_g, ln_b, stats, N);
    bn_finalize_kernel<<<1, 64, 0, stream>>>(stats, bn_g, bn_b, N);
    ln_bn_apply_kernel<<<1024, 256, 0, stream>>>(
        X, ln_g, ln_b, stats, X, X0, N, 0);

    // ---- 3 propagation layers ----
    const int ncopy4 = N * DD / 4;
    for (int l = 0; l < 3; ++l) {
        copy4_kernel<<<2048, 256, 0, stream>>>(
            (const float4*)X0, (float4*)AGG, ncopy4);
        spmm_kernel<<<8192, 256, 0, stream>>>(erow, ecol, evalv, X, AGG, NNZ);

        const float* lg = ln_g + (size_t)(l + 1) * DD;
        const float* lb = ln_b + (size_t)(l + 1) * DD;
        const float* bg = bn_g + (size_t)(l + 1) * DD;
        const float* bb = bn_b + (size_t)(l + 1) * DD;
        zero_stats_kernel<<<1, 256, 0, stream>>>(stats);
        ln_stats_kernel<<<1024, 256, 0, stream>>>(AGG, lg, lb, stats, N);
        bn_finalize_kernel<<<1, 64, 0, stream>>>(stats, bg, bb, N);
        ln_bn_apply_kernel<<<1024, 256, 0, stream>>>(
            AGG, lg, lb, stats, X, nullptr, N, (l < 2) ? 1 : 0);
    }
    (void)in_sizes; (void)n_in; (void)out_size; (void)ws_size;
}